// GNN_28286654612096
// MI455X (gfx1250) — compile-verified
//
#include <hip/hip_runtime.h>
#include <math.h>

// ---------------- problem constants ----------------
#define NPTS   65536
#define PGRAPH 1024
#define NGRAPH 64
#define KNN    16
#define HID    128
#define CONCAT 384
#define OUTD   40
#define BN_SC  0.9999950000374997f   // 1/sqrt(1+1e-5)

typedef __attribute__((ext_vector_type(2))) float v2f;
typedef __attribute__((ext_vector_type(8))) float v8f;
typedef __attribute__((ext_vector_type(4))) int   v4i;
typedef __attribute__((address_space(1))) v4i* gptr_v4i;   // global (AS1) int4*
typedef __attribute__((address_space(3))) v4i* lptr_v4i;   // LDS (AS3) int4*

// f32 WMMA: D(16x16,f32) = A(16x4,f32) * B(4x16,f32) + C
__device__ __forceinline__ v8f wmma_f32(v2f a, v2f b, v8f c) {
  return __builtin_amdgcn_wmma_f32_16x16x4_f32(false, a, false, b, (short)0, c,
                                               false, false);
}

// ---- CDNA5 async global->LDS copy (ASYNCcnt-tracked), with safe fallback ----
__device__ __forceinline__ void async_copy_f4(float* lds_dst, const float* gsrc) {
#if __has_builtin(__builtin_amdgcn_global_load_async_to_lds_b128)
  // signature (per hipcc diagnostic): (v4i AS1* src, v4i AS3* dst, imm off, imm cpol)
  __builtin_amdgcn_global_load_async_to_lds_b128(
      (gptr_v4i)(gsrc), (lptr_v4i)(lds_dst), 0, 0);
#else
  *(float4*)lds_dst = *(const float4*)gsrc;   // sync fallback, still LDS-staged
#endif
}

__device__ __forceinline__ void wait_async0() {
#if __has_builtin(__builtin_amdgcn_s_wait_asynccnt)
  __builtin_amdgcn_s_wait_asynccnt(0);
#else
  asm volatile("s_wait_asynccnt 0x0" ::: "memory");
#endif
}

// ---------------- small utility kernels ----------------
__global__ void pad_x_kernel(const float* __restrict__ x, float* __restrict__ xp) {
  int i = blockIdx.x * blockDim.x + threadIdx.x;
  if (i >= NPTS) return;
  xp[i * 4 + 0] = x[i * 3 + 0];
  xp[i * 4 + 1] = x[i * 3 + 1];
  xp[i * 4 + 2] = x[i * 3 + 2];
  xp[i * 4 + 3] = 0.0f;
}

__global__ void pad_w0_kernel(const float* __restrict__ w, float* __restrict__ wp) {
  int i = blockIdx.x * blockDim.x + threadIdx.x;
  if (i >= 4 * HID) return;
  int r = i >> 7, c = i & 127;
  wp[i] = (r < 3) ? w[r * HID + c] : 0.0f;
}

__global__ void sqnorm_kernel(const float* __restrict__ F, int lda, int D,
                              float* __restrict__ sq) {
  int i = blockIdx.x * blockDim.x + threadIdx.x;
  if (i >= NPTS) return;
  const float* r = F + (size_t)i * lda;
  float s = 0.0f;
  for (int f = 0; f < D; ++f) { float v = r[f]; s += v * v; }
  sq[i] = s;
}

// ---------------- kNN: WMMA Gram tiles + per-lane top-16 ----------------
// One wave owns 32 consecutive query rows. A fragments (the wave's 32 query
// rows) are hoisted to registers and reused across all 64 column tiles, so the
// inner loop is 1 global b64 load per 2 WMMAs. Selection key = sq[j] - 2*dot
// (row-constant sq[i] dropped; cannot change per-row top-k order).
template <int D>
__global__ void knn_wmma_kernel(const float* __restrict__ F, int lda,
                                const float* __restrict__ sq,
                                int* __restrict__ idx_out) {
  constexpr int NC = D / 4;
  __shared__ float smem[8][32 * 17];   // 8 waves/block, stride 17 (conflict-free)
  const int lane = threadIdx.x & 31;
  const int wid  = threadIdx.x >> 5;
  const int n    = lane & 15;
  const int h    = lane >> 4;
  const int wgid = blockIdx.x * (blockDim.x >> 5) + wid;
  const int qbase = wgid * 32;
  const int graph = qbase >> 10;
  const int pbase = graph << 10;

  float kv[KNN]; int ki[KNN];
#pragma unroll
  for (int t = 0; t < KNN; ++t) { kv[t] = 3.4e38f; ki[t] = qbase + lane; }

  // hoist A fragments (32-bit A 16x4 lane layout: lane m<16 -> K=k0..k0+1,
  // lane m>=16 -> K=k0+2..k0+3)
  const size_t qa0 = (size_t)(qbase + n) * lda;
  const size_t qa1 = (size_t)(qbase + 16 + n) * lda;
  v2f a0[NC], a1[NC];
#pragma unroll
  for (int kc = 0; kc < NC; ++kc) {
    const int ko = kc * 4 + 2 * h;
    a0[kc] = *(const v2f*)(F + qa0 + ko);
    a1[kc] = *(const v2f*)(F + qa1 + ko);
  }

  for (int ct = 0; ct < PGRAPH / 16; ++ct) {
    v8f acc0 = {}; v8f acc1 = {};
    const size_t pb = (size_t)(pbase + ct * 16 + n) * lda;
#pragma unroll
    for (int kc = 0; kc < NC; ++kc) {
      const int ko = kc * 4 + 2 * h;   // 32-bit B 4x16 lane layout
      v2f b = *(const v2f*)(F + pb + ko);
      acc0 = wmma_f32(a0[kc], b, acc0);
      acc1 = wmma_f32(a1[kc], b, acc1);
    }
    // stage 32x16 dot tile: C/D layout -> row = j + 8*h (+16 for second tile)
    float* s = &smem[wid][0];
#pragma unroll
    for (int j = 0; j < 8; ++j) {
      s[(j + 8 * h) * 17 + n]      = acc0[j];
      s[(16 + j + 8 * h) * 17 + n] = acc1[j];
    }
    // same-wave DS readback (compiler inserts s_wait_dscnt)
    const float* srow = &smem[wid][lane * 17];
    const int cg0 = pbase + ct * 16;
#pragma unroll
    for (int c = 0; c < 16; ++c) {
      float key = sq[cg0 + c] - 2.0f * srow[c];
      int  cidx = cg0 + c;
      if (key < kv[KNN - 1]) {        // fast reject against current worst
#pragma unroll
        for (int t = 0; t < KNN; ++t) {
          bool lt  = key < kv[t];
          float tv = kv[t]; int ti = ki[t];
          kv[t] = lt ? key  : tv;  ki[t] = lt ? cidx : ti;
          key   = lt ? tv   : key; cidx  = lt ? ti   : cidx;
        }
      }
    }
  }
  int* o = idx_out + (size_t)(qbase + lane) * KNN;
#pragma unroll
  for (int t = 0; t < KNN; ++t) o[t] = ki[t];
}

// ---------------- degree / aggregation ----------------
__global__ void deg_init_kernel(float* __restrict__ deg) {
  int i = blockIdx.x * blockDim.x + threadIdx.x;
  if (i < NPTS) deg[i] = 1.0f;                    // self loop
}

__global__ void deg_count_kernel(const int* __restrict__ idx, float* __restrict__ deg) {
  int e = blockIdx.x * blockDim.x + threadIdx.x;
  if (e < NPTS * KNN) atomicAdd(&deg[idx[e]], 1.0f);
}

__global__ void deg_rsqrt_kernel(float* __restrict__ deg) {
  int i = blockIdx.x * blockDim.x + threadIdx.x;
  if (i < NPTS) deg[i] = rsqrtf(deg[i]);
}

__global__ void scale_init_kernel(const float* __restrict__ h, const float* __restrict__ dis,
                                  float* __restrict__ u, float* __restrict__ acc) {
  int t = blockIdx.x * blockDim.x + threadIdx.x;   // one float4 per thread
  if (t >= NPTS * (HID / 4)) return;
  int i = t >> 5;                                  // 32 float4 per row
  float d = dis[i];
  float4 v = ((const float4*)h)[t];
  v.x *= d; v.y *= d; v.z *= d; v.w *= d;
  ((float4*)u)[t]   = v;
  ((float4*)acc)[t] = v;                           // self-loop contribution
}

__global__ void scatter_kernel(const int* __restrict__ idx, const float* __restrict__ u,
                               float* __restrict__ acc) {
  long long t = (long long)blockIdx.x * blockDim.x + threadIdx.x;
  if (t >= (long long)NPTS * KNN * (HID / 4)) return;
  int f4 = (int)(t & 31);
  int e  = (int)(t >> 5);
  int r  = e >> 4;                 // query node
  int c  = idx[e];                 // target node (global id)
  float4 v = ((const float4*)u)[r * 32 + f4];
  float* d = acc + (size_t)c * HID + f4 * 4;
  atomicAdd(d + 0, v.x); atomicAdd(d + 1, v.y);
  atomicAdd(d + 2, v.z); atomicAdd(d + 3, v.w);
}

__global__ void agg_finalize_kernel(const float* __restrict__ acc, const float* __restrict__ dis,
                                    const float* __restrict__ bias, float* __restrict__ out,
                                    int ldc) {
  int t = blockIdx.x * blockDim.x + threadIdx.x;
  if (t >= NPTS * HID) return;
  int i = t >> 7, f = t & 127;
  float v = dis[i] * acc[t] + bias[f];
  v = v > 0.0f ? v : (__expf(v) - 1.0f);          // ELU
  out[(size_t)i * ldc + f] = v;
}

// ---------------- f32-WMMA GEMM with async-LDS weight staging ----------------
// 4 waves/block, each wave: 16 rows x NT*16 cols. B is staged through LDS in
// double-buffered 32xldb tiles via GLOBAL_LOAD_ASYNC_TO_LDS_B128; the
// s_wait_asynccnt lands after the compute phase so DMA overlaps WMMA work.
// mode: 0=raw, 1=+bias, 2=+bias->ReLU->BN.
template <int NT>
__global__ void wmma_gemm_kernel(const float* __restrict__ A, int lda,
                                 const float* __restrict__ B, int ldb, int ncols,
                                 float* __restrict__ C, int ldc, int K,
                                 const float* __restrict__ bias,
                                 const float* __restrict__ bng,
                                 const float* __restrict__ bnb, int mode) {
  __shared__ float bsm[2][32 * HID];               // 32 KB
  const int lane  = threadIdx.x & 31;
  const int wid   = threadIdx.x >> 5;
  const int n     = lane & 15;
  const int h     = lane >> 4;
  const int mtile = blockIdx.x * (blockDim.x >> 5) + wid;
  const size_t abase = (size_t)(mtile * 16 + n) * lda;

  v8f acc[NT];
  v8f zero = {};
#pragma unroll
  for (int i = 0; i < NT; ++i) acc[i] = zero;

  const int nkt = (K + 31) >> 5;                   // 32-row k-tiles
  const int n4  = (32 * ldb) >> 2;                 // float4s per tile

  // prologue: stage tile 0
  for (int t = threadIdx.x; t < n4; t += blockDim.x)
    async_copy_f4(&bsm[0][t << 2], B + ((size_t)t << 2));
  wait_async0();
  __syncthreads();

  for (int kt = 0; kt < nkt; ++kt) {
    const int cur = kt & 1;
    if (kt + 1 < nkt) {                            // prefetch next tile (async)
      const float* src = B + (size_t)(kt + 1) * 32 * ldb;
      for (int t = threadIdx.x; t < n4; t += blockDim.x)
        async_copy_f4(&bsm[cur ^ 1][t << 2], src + ((size_t)t << 2));
    }
    const float* bt = &bsm[cur][0];
#pragma unroll
    for (int kc = 0; kc < 8; ++kc) {
      const int k0 = kt * 32 + kc * 4;
      if (k0 >= K) break;                          // partial last tile (K=4)
      const int ko = kc * 4 + 2 * h;
      v2f a = *(const v2f*)(A + abase + kt * 32 + ko);
#pragma unroll
      for (int nt = 0; nt < NT; ++nt) {
        const int col = nt * 16 + n;
        v2f b;
        if (col < ncols) {
          b.x = bt[ko * ldb + col];
          b.y = bt[(ko + 1) * ldb + col];
        } else { b.x = 0.0f; b.y = 0.0f; }
        acc[nt] = wmma_f32(a, b, acc[nt]);
      }
    }
    if (kt + 1 < nkt) wait_async0();               // next tile landed
    __syncthreads();                               // safe to overwrite old buf
  }

#pragma unroll
  for (int nt = 0; nt < NT; ++nt) {
    const int col = nt * 16 + n;
    if (col >= ncols) continue;
    float bs = (mode >= 1) ? bias[col] : 0.0f;
    float g  = (mode == 2) ? bng[col] * BN_SC : 1.0f;
    float bb = (mode == 2) ? bnb[col] : 0.0f;
#pragma unroll
    for (int j = 0; j < 8; ++j) {
      float v = acc[nt][j] + bs;
      if (mode == 2) v = fmaxf(v, 0.0f) * g + bb;
      C[(size_t)(mtile * 16 + j + 8 * h) * ldc + col] = v;
    }
  }
}

// ---------------- pool & log-softmax ----------------
__global__ void pool_kernel(const float* __restrict__ z, float* __restrict__ g) {
  int t = blockIdx.x * blockDim.x + threadIdx.x;
  if (t >= NGRAPH * HID) return;
  int b = t >> 7, f = t & 127;
  const float* p = z + (size_t)b * PGRAPH * HID + f;
  float s = 0.0f;
  for (int r = 0; r < PGRAPH; ++r) s += p[(size_t)r * HID];
  g[t] = s;
}

__global__ void logsoftmax_kernel(const float* __restrict__ g, float* __restrict__ out) {
  int r = blockIdx.x * blockDim.x + threadIdx.x;
  if (r >= NGRAPH) return;
  const float* x = g + r * OUTD;
  float mx = -3.4e38f;
  for (int c = 0; c < OUTD; ++c) mx = fmaxf(mx, x[c]);
  float s = 0.0f;
  for (int c = 0; c < OUTD; ++c) s += __expf(x[c] - mx);
  float ls = __logf(s) + mx;
  for (int c = 0; c < OUTD; ++c) out[r * OUTD + c] = x[c] - ls;
}

// ---------------- host-side helpers ----------------
static void run_gcn_aggregation(hipStream_t stream, const int* IDX, const float* H,
                                float* DIS, float* U, float* ACC,
                                const float* bias, float* out_col, int ldc) {
  deg_init_kernel<<<NPTS / 256, 256, 0, stream>>>(DIS);
  deg_count_kernel<<<(NPTS * KNN) / 256, 256, 0, stream>>>(IDX, DIS);
  deg_rsqrt_kernel<<<NPTS / 256, 256, 0, stream>>>(DIS);
  scale_init_kernel<<<(NPTS * (HID / 4)) / 256, 256, 0, stream>>>(H, DIS, U, ACC);
  scatter_kernel<<<(unsigned)(((long long)NPTS * KNN * (HID / 4)) / 256), 256, 0, stream>>>(IDX, U, ACC);
  agg_finalize_kernel<<<(NPTS * HID) / 256, 256, 0, stream>>>(ACC, DIS, bias, out_col, ldc);
}

extern "C" void kernel_launch(void* const* d_in, const int* in_sizes, int n_in,
                              void* d_out, int out_size, void* d_ws, size_t ws_size,
                              hipStream_t stream) {
  const float* x    = (const float*)d_in[0];
  const float* W0   = (const float*)d_in[1];
  const float* b0   = (const float*)d_in[2];
  const float* Wl   = (const float*)d_in[3];
  const float* bl   = (const float*)d_in[4];
  const float* p1w  = (const float*)d_in[5];
  const float* p1b  = (const float*)d_in[6];
  const float* bn1g = (const float*)d_in[7];
  const float* bn1b = (const float*)d_in[8];
  const float* p2w  = (const float*)d_in[9];
  const float* p2b  = (const float*)d_in[10];
  const float* bn2g = (const float*)d_in[11];
  const float* bn2b = (const float*)d_in[12];
  const float* m1w  = (const float*)d_in[13];
  const float* m1b  = (const float*)d_in[14];
  const float* bn3g = (const float*)d_in[15];
  const float* bn3b = (const float*)d_in[16];
  const float* m2w  = (const float*)d_in[17];
  const float* m2b  = (const float*)d_in[18];
  const float* bn4g = (const float*)d_in[19];
  const float* bn4b = (const float*)d_in[20];
  const float* m3w  = (const float*)d_in[21];
  const float* m3b  = (const float*)d_in[22];
  // d_in[23] = batch (int64): unused, graphs are contiguous blocks of 1024

  // workspace carve (~198 MB total)
  char* ws = (char*)d_ws;
  size_t off = 0;
  auto carve = [&](size_t bytes) -> void* {
    void* p = ws + off;
    off = (off + bytes + 255) & ~(size_t)255;
    return p;
  };
  float* XS   = (float*)carve((size_t)NPTS * CONCAT * 4);  // [N,384] concat h1|h2|h3
  float* TMP1 = (float*)carve((size_t)NPTS * HID * 4);     // h = x@W / z1
  float* TMP2 = (float*)carve((size_t)NPTS * HID * 4);     // u = h*dis / z2
  float* ACC  = (float*)carve((size_t)NPTS * HID * 4);     // scatter accumulator
  float* XPAD = (float*)carve((size_t)NPTS * 4 * 4);       // x padded to 4 cols
  float* SQ   = (float*)carve((size_t)NPTS * 4);
  float* DIS  = (float*)carve((size_t)NPTS * 4);
  int*   IDX  = (int*)  carve((size_t)NPTS * KNN * 4);
  float* W0P  = (float*)carve(4 * HID * 4);
  float* G0   = (float*)carve(NGRAPH * HID * 4);
  float* G1   = (float*)carve(NGRAPH * HID * 4);
  float* G2   = (float*)carve(NGRAPH * HID * 4);
  float* G3   = (float*)carve(NGRAPH * OUTD * 4);

  const int GEMM_BLK  = 128;                     // 4 waves -> 64 rows/block
  const int GEMM_GRID = NPTS / 64;               // 1024 blocks for N-row GEMMs
  const int KNN_GRID  = (NPTS / 32) / 8;         // 8 waves/block, 32 queries/wave

  // ---- layer 0: kNN on raw xyz, GCNConv(3->128) ----
  pad_x_kernel<<<NPTS / 256, 256, 0, stream>>>(x, XPAD);
  pad_w0_kernel<<<2, 256, 0, stream>>>(W0, W0P);
  sqnorm_kernel<<<NPTS / 256, 256, 0, stream>>>(XPAD, 4, 3, SQ);
  knn_wmma_kernel<4><<<KNN_GRID, 256, 0, stream>>>(XPAD, 4, SQ, IDX);
  wmma_gemm_kernel<8><<<GEMM_GRID, GEMM_BLK, 0, stream>>>(
      XPAD, 4, W0P, HID, HID, TMP1, HID, 4, nullptr, nullptr, nullptr, 0);
  run_gcn_aggregation(stream, IDX, TMP1, DIS, TMP2, ACC, b0, XS + 0, CONCAT);

  // ---- layers 1..2: dynamic kNN on current 128-d features ----
  for (int l = 0; l < 2; ++l) {
    const float* hprev = XS + (size_t)l * HID;   // lda = 384 column slice
    sqnorm_kernel<<<NPTS / 256, 256, 0, stream>>>(hprev, CONCAT, HID, SQ);
    knn_wmma_kernel<128><<<KNN_GRID, 256, 0, stream>>>(hprev, CONCAT, SQ, IDX);
    wmma_gemm_kernel<8><<<GEMM_GRID, GEMM_BLK, 0, stream>>>(
        hprev, CONCAT, Wl + (size_t)l * HID * HID, HID, HID, TMP1, HID, HID,
        nullptr, nullptr, nullptr, 0);
    run_gcn_aggregation(stream, IDX, TMP1, DIS, TMP2, ACC, bl + (size_t)l * HID,
                        XS + (size_t)(l + 1) * HID, CONCAT);
  }

  // ---- projection head: [N,384]@[384,128] and [N,128]@[128,128], ReLU+BN ----
  wmma_gemm_kernel<8><<<GEMM_GRID, GEMM_BLK, 0, stream>>>(
      XS, CONCAT, p1w, HID, HID, TMP1, HID, CONCAT, p1b, bn1g, bn1b, 2);
  wmma_gemm_kernel<8><<<GEMM_GRID, GEMM_BLK, 0, stream>>>(
      TMP1, HID, p2w, HID, HID, TMP2, HID, HID, p2b, bn2g, bn2b, 2);

  // ---- global_add_pool -> [64,128] ----
  pool_kernel<<<(NGRAPH * HID) / 256, 256, 0, stream>>>(TMP2, G0);

  // ---- MLP head (M=64 rows: one block of 4 waves) ----
  wmma_gemm_kernel<8><<<1, GEMM_BLK, 0, stream>>>(
      G0, HID, m1w, HID, HID, G1, HID, HID, m1b, bn3g, bn3b, 2);
  wmma_gemm_kernel<8><<<1, GEMM_BLK, 0, stream>>>(
      G1, HID, m2w, HID, HID, G2, HID, HID, m2b, bn4g, bn4b, 2);
  wmma_gemm_kernel<3><<<1, GEMM_BLK, 0, stream>>>(
      G2, HID, m3w, OUTD, OUTD, G3, OUTD, HID, m3b, nullptr, nullptr, 1);

  logsoftmax_kernel<<<1, 64, 0, stream>>>(G3, (float*)d_out);
}